// ngcf3_88957362635438
// MI455X (gfx1250) — compile-verified
//
#include <hip/hip_runtime.h>
#include <hip/hip_bf16.h>

// ---------------------------------------------------------------------------
// NGCF-3 pipeline for MI455X (gfx1250, wave32).
// Dense GEMMs use V_WMMA_F32_16X16X4_F32 (fp32 matrix path, matches the fp32
// reference; workload is L2-resident so SpMM atomics dominate, not GEMM).
// ---------------------------------------------------------------------------

typedef __attribute__((ext_vector_type(2))) float v2f;
typedef __attribute__((ext_vector_type(8))) float v8f;

#define S_CNT   50000
#define I_CNT   50000
#define N_CNT   100000
#define D_DIM   128
#define EA_CNT  1600000
#define ES_CNT  1000000
#define B_CNT   1024
#define Q_CNT   10000

// ---------------------------------------------------------------------------
// SpMM: Y[row] += val * X[col]  (COO, unsorted -> atomic scatter-add).
// One wave (32 lanes) per edge; float4 per lane covers D=128.
// ---------------------------------------------------------------------------
__global__ __launch_bounds__(256) void spmm_atomic(
    const int* __restrict__ rows, const int* __restrict__ cols,
    const float* __restrict__ vals, const float* __restrict__ X,
    float* __restrict__ Y, int nnz) {
  int e = blockIdx.x * 8 + (threadIdx.x >> 5);
  if (e >= nnz) return;
  int lane = threadIdx.x & 31;
  int row = rows[e];
  int col = cols[e];
  float v = vals[e];
  const float4* xs = (const float4*)(X + (size_t)col * D_DIM);
  float4 d = xs[lane];
  float* yb = Y + (size_t)row * D_DIM + lane * 4;
  atomicAdd(yb + 0, v * d.x);
  atomicAdd(yb + 1, v * d.y);
  atomicAdd(yb + 2, v * d.z);
  atomicAdd(yb + 3, v * d.w);
}

// ---------------------------------------------------------------------------
// Dense GEMM: Y[N x 128] = X[N x 128] @ W[128 x 128] via fp32 WMMA.
// grid.x = N/16 row tiles; 8 waves per block each own one 16-col tile.
// A frag (16x4 f32): lane L -> A[L%16][k0 + 2*(L/16) + v], v in {0,1}
// B frag (4x16 f32): lane L -> W[k0 + 2*(L/16) + v][n0 + L%16]
// C/D (16x16 f32):   acc[r] at lane L -> C[r + 8*(L/16)][L%16]
// ---------------------------------------------------------------------------
__global__ __launch_bounds__(256) void gemm_xw(
    const float* __restrict__ X, const float* __restrict__ W,
    float* __restrict__ Y) {
  int wave = threadIdx.x >> 5;
  int lane = threadIdx.x & 31;
  int m0 = blockIdx.x * 16;
  int n0 = wave * 16;
  int r16 = lane & 15;     // m for A, n for B/C
  int kh = (lane >> 4) * 2;

  const float* Arow = X + (size_t)(m0 + r16) * D_DIM;
  const float* Bcol = W + n0 + r16;

  v8f acc = {};
#pragma unroll
  for (int k0 = 0; k0 < D_DIM; k0 += 4) {
    v2f a, b;
    a.x = Arow[k0 + kh + 0];
    a.y = Arow[k0 + kh + 1];
    b.x = Bcol[(size_t)(k0 + kh + 0) * D_DIM];
    b.y = Bcol[(size_t)(k0 + kh + 1) * D_DIM];
    acc = __builtin_amdgcn_wmma_f32_16x16x4_f32(
        false, a, false, b, (short)0, acc, false, false);
  }

  int mh = (lane >> 4) * 8;
  float* Yb = Y + (size_t)m0 * D_DIM + n0;
#pragma unroll
  for (int r = 0; r < 8; ++r)
    Yb[(size_t)(mh + r) * D_DIM + r16] = acc[r];
}

// ---------------------------------------------------------------------------
// Fused bias (+ReLU +row L2-normalize). One wave per row, float4 per lane,
// wave32 shuffle-xor reduction for the squared norm.
// ---------------------------------------------------------------------------
__global__ __launch_bounds__(256) void bias_act(
    float* __restrict__ H, const float* __restrict__ bias,
    int n_rows, int do_relu, int do_norm) {
  int row = blockIdx.x * 8 + (threadIdx.x >> 5);
  if (row >= n_rows) return;
  int lane = threadIdx.x & 31;
  float4* hp = (float4*)(H + (size_t)row * D_DIM);
  float4 h = hp[lane];
  float4 bb = ((const float4*)bias)[lane];
  h.x += bb.x; h.y += bb.y; h.z += bb.z; h.w += bb.w;
  if (do_relu) {
    h.x = fmaxf(h.x, 0.0f); h.y = fmaxf(h.y, 0.0f);
    h.z = fmaxf(h.z, 0.0f); h.w = fmaxf(h.w, 0.0f);
  }
  if (do_norm) {
    float ss = h.x * h.x + h.y * h.y + h.z * h.z + h.w * h.w;
#pragma unroll
    for (int off = 16; off > 0; off >>= 1)
      ss += __shfl_xor(ss, off, 32);
    float inv = 1.0f / fmaxf(sqrtf(ss), 1e-12f);
    h.x *= inv; h.y *= inv; h.z *= inv; h.w *= inv;
  }
  hp[lane] = h;
}

// ---------------------------------------------------------------------------
// Output GEMM: out[B x Q] = H[batch] @ H[items]^T via fp32 WMMA.
// One wave per 16x16 tile; B^T gather uses the same lane pattern as A.
// 64 x 625 = 40000 tiles, 8 waves/block -> 5000 blocks (exact).
// ---------------------------------------------------------------------------
__global__ __launch_bounds__(256) void out_gemm(
    const float* __restrict__ H, const int* __restrict__ bidx,
    const int* __restrict__ qidx, float* __restrict__ out) {
  const int QT = Q_CNT / 16;   // 625
  int tile = blockIdx.x * 8 + (threadIdx.x >> 5);
  int ti = tile / QT;
  int tj = tile % QT;
  int lane = threadIdx.x & 31;
  int r16 = lane & 15;
  int kh = (lane >> 4) * 2;

  const float* Ar = H + (size_t)bidx[ti * 16 + r16] * D_DIM;
  const float* Br = H + (size_t)qidx[tj * 16 + r16] * D_DIM;

  v8f acc = {};
#pragma unroll
  for (int k0 = 0; k0 < D_DIM; k0 += 4) {
    v2f a, b;
    a.x = Ar[k0 + kh + 0];
    a.y = Ar[k0 + kh + 1];
    b.x = Br[k0 + kh + 0];
    b.y = Br[k0 + kh + 1];
    acc = __builtin_amdgcn_wmma_f32_16x16x4_f32(
        false, a, false, b, (short)0, acc, false, false);
  }

  int mh = (lane >> 4) * 8;
  float* ob = out + (size_t)(ti * 16) * Q_CNT + tj * 16;
#pragma unroll
  for (int r = 0; r < 8; ++r)
    ob[(size_t)(mh + r) * Q_CNT + r16] = acc[r];
}

// ---------------------------------------------------------------------------
extern "C" void kernel_launch(void* const* d_in, const int* in_sizes, int n_in,
                              void* d_out, int out_size, void* d_ws, size_t ws_size,
                              hipStream_t stream) {
  (void)in_sizes; (void)n_in; (void)out_size; (void)ws_size;

  const float* item_emb = (const float*)d_in[0];
  const float* W1 = (const float*)d_in[1];
  const float* b1 = (const float*)d_in[2];
  const float* W2 = (const float*)d_in[3];
  const float* b2 = (const float*)d_in[4];
  const float* W3 = (const float*)d_in[5];
  const float* b3 = (const float*)d_in[6];
  const int*   A_rows = (const int*)d_in[7];
  const int*   A_cols = (const int*)d_in[8];
  const float* A_vals = (const float*)d_in[9];
  const int*   s_rows = (const int*)d_in[10];
  const int*   s_cols = (const int*)d_in[11];
  const float* s_vals = (const float*)d_in[12];
  const int*   bidx   = (const int*)d_in[13];
  const int*   qidx   = (const int*)d_in[14];

  // Workspace: two N x D fp32 buffers (ping-pong; X is dead after X@W).
  float* xbuf = (float*)d_ws;
  float* tbuf = xbuf + (size_t)N_CNT * D_DIM;

  // x = concat(spmm(sess, item_emb), item_emb)
  hipMemsetAsync(xbuf, 0, (size_t)S_CNT * D_DIM * sizeof(float), stream);
  hipMemcpyAsync(xbuf + (size_t)S_CNT * D_DIM, item_emb,
                 (size_t)I_CNT * D_DIM * sizeof(float),
                 hipMemcpyDeviceToDevice, stream);
  spmm_atomic<<<ES_CNT / 8, 256, 0, stream>>>(s_rows, s_cols, s_vals,
                                              item_emb, xbuf, ES_CNT);

  const float* Ws[3] = {W1, W2, W3};
  const float* bs[3] = {b1, b2, b3};
  for (int l = 0; l < 3; ++l) {
    // tbuf = xbuf @ W_l      (WMMA fp32)
    gemm_xw<<<N_CNT / 16, 256, 0, stream>>>(xbuf, Ws[l], tbuf);
    // xbuf = spmm(A, tbuf)   (xbuf is free after the GEMM)
    hipMemsetAsync(xbuf, 0, (size_t)N_CNT * D_DIM * sizeof(float), stream);
    spmm_atomic<<<EA_CNT / 8, 256, 0, stream>>>(A_rows, A_cols, A_vals,
                                                tbuf, xbuf, EA_CNT);
    // + bias, relu+l2norm on layers 0,1; plain bias on layer 2 (h3)
    bias_act<<<N_CNT / 8, 256, 0, stream>>>(xbuf, bs[l], N_CNT,
                                            l < 2 ? 1 : 0, l < 2 ? 1 : 0);
  }

  // out = h3[batch] @ h3[items]^T   (WMMA fp32)
  out_gemm<<<(B_CNT / 16) * (Q_CNT / 16) / 8, 256, 0, stream>>>(
      xbuf, bidx, qidx, (float*)d_out);
}